// SAM3E_27934467293448
// MI455X (gfx1250) — compile-verified
//
#include <hip/hip_runtime.h>

// out[b,n,d] = F[b,n,d] * ( (F Kw F^T) @ F )[b,n,d] + (F @ Qw^T)[b,n,d]
// B=8192, N=50 (padded to 64), D=64. All fp32, exact via V_WMMA_F32_16X16X4_F32.

typedef float v2f __attribute__((ext_vector_type(2)));
typedef float v8f __attribute__((ext_vector_type(8)));

#define NROWS 50
#define DDIM  64
#define PITCH 68            // floats; 68%64=4 -> conflict-free row-pair loads
#define REG   (64 * PITCH)  // one 64-row LDS panel

// A-operand / transposed-B-operand fragment load: lane (l16, half) reads the
// two consecutive K elements {kb+2*half, kb+2*half+1} of row `row`.
// Matches ISA 16x4 A layout: K = 2*(lane/16) + vgpr_index.
__device__ __forceinline__ v2f ld2(const float* s, int row, int col) {
  return *(const v2f*)(s + row * PITCH + col);   // 8B aligned: PITCH even, col even
}

__device__ __forceinline__ v8f wmma_f32(v2f a, v2f b, v8f c) {
  return __builtin_amdgcn_wmma_f32_16x16x4_f32(
      /*neg_a=*/false, a, /*neg_b=*/false, b,
      /*c_mod=*/(short)0, c, /*reuse_a=*/false, /*reuse_b=*/false);
}

__global__ __launch_bounds__(128) void SAM3E_27934467293448_kernel(
    const float* __restrict__ F, const float* __restrict__ Kw,
    const float* __restrict__ Qw, float* __restrict__ out) {
  __shared__ float smem[3 * REG];
  float* sF  = smem;            // F panel (64x64, rows 50..63 zeroed)
  float* sKw = smem + REG;      // Kw, later reused as S
  float* sQw = smem + 2 * REG;  // Qw, later reused as KF
  float* sKF = sQw;
  float* sS  = sKw;

  const int b   = blockIdx.x;
  const int tid = threadIdx.x;
  const float* Fb = F + (size_t)b * (NROWS * DDIM);

  // ---- Fill LDS: F (50x64, float4-coalesced), zero pad, Kw & Qw (64x64) ----
  for (int i = tid; i < (NROWS * DDIM) / 4; i += 128) {
    float4 v = ((const float4*)Fb)[i];
    int r = i >> 4, c4 = (i & 15) << 2;          // 16 float4 per row
    *(float4*)(sF + r * PITCH + c4) = v;
  }
  for (int i = tid; i < (64 - NROWS) * DDIM; i += 128) {
    int r = NROWS + i / DDIM, c = i % DDIM;
    sF[r * PITCH + c] = 0.0f;
  }
  for (int i = tid; i < (DDIM * DDIM) / 4; i += 128) {
    int r = i >> 4, c4 = (i & 15) << 2;
    *(float4*)(sKw + r * PITCH + c4) = ((const float4*)Kw)[i];
    *(float4*)(sQw + r * PITCH + c4) = ((const float4*)Qw)[i];
  }
  __syncthreads();

  const int lane = tid & 31;
  const int wv   = tid >> 5;      // wave owns M rows [16*wv, 16*wv+16)
  const int l16  = lane & 15;
  const int half = lane >> 4;
  const int mrow = wv * 16 + l16; // A-fragment row for this lane
  const int kofs = 2 * half;      // K sub-offset per ISA A/B layout

  // ---- Stage A1: Qacc = F @ Qw^T, kept in registers (4 N-tiles x v8f) ----
  v8f qacc[4];
#pragma unroll
  for (int nt = 0; nt < 4; ++nt) {
    v8f c = {};
#pragma unroll
    for (int kb = 0; kb < DDIM; kb += 4) {
      v2f a = ld2(sF,  mrow,          kb + kofs);
      v2f w = ld2(sQw, nt * 16 + l16, kb + kofs);  // B = Qw^T: same load shape
      c = wmma_f32(a, w, c);
    }
    qacc[nt] = c;
  }
  __syncthreads();  // all Qw reads done -> sQw region reusable as sKF

  // ---- Stage A2: KF = F @ Kw^T -> sKF (own 16 rows; rows>=50 are zero) ----
#pragma unroll
  for (int nt = 0; nt < 4; ++nt) {
    v8f c = {};
#pragma unroll
    for (int kb = 0; kb < DDIM; kb += 4) {
      v2f a = ld2(sF,  mrow,          kb + kofs);
      v2f w = ld2(sKw, nt * 16 + l16, kb + kofs);  // B = Kw^T
      c = wmma_f32(a, w, c);
    }
#pragma unroll
    for (int r = 0; r < 8; ++r)   // C/D layout: (M = r + 8*half, N = l16)
      sKF[(wv * 16 + r + 8 * half) * PITCH + nt * 16 + l16] = c[r];
  }
  __syncthreads();  // KF visible to all waves; sKw region now dead -> sS

  // ---- Stage B: S = F @ KF^T -> sS (own 16 rows) ----
#pragma unroll
  for (int nt = 0; nt < 4; ++nt) {
    v8f c = {};
#pragma unroll
    for (int kb = 0; kb < DDIM; kb += 4) {
      v2f a = ld2(sF,  mrow,          kb + kofs);
      v2f k = ld2(sKF, nt * 16 + l16, kb + kofs);  // B = KF^T
      c = wmma_f32(a, k, c);
    }
#pragma unroll
    for (int r = 0; r < 8; ++r)
      sS[(wv * 16 + r + 8 * half) * PITCH + nt * 16 + l16] = c[r];
  }
  // No barrier: stage C consumes only this wave's own S rows (LDS in-order).

  // ---- Stage C: T = S @ F ; out = F*T + Qacc ----
  float* outb = out + (size_t)b * (NROWS * DDIM);
#pragma unroll
  for (int nt = 0; nt < 4; ++nt) {
    v8f c = {};
#pragma unroll
    for (int kb = 0; kb < DDIM; kb += 4) {
      v2f a = ld2(sS, mrow, kb + kofs);            // A = S rows (K = m)
      int krow = kb + kofs;                        // B = F (non-transposed)
      v2f bv;
      bv[0] = sF[(krow    ) * PITCH + nt * 16 + l16];
      bv[1] = sF[(krow + 1) * PITCH + nt * 16 + l16];
      c = wmma_f32(a, bv, c);
    }
#pragma unroll
    for (int r = 0; r < 8; ++r) {
      int n = wv * 16 + r + 8 * half;
      int d = nt * 16 + l16;
      if (n < NROWS) {
        float f = sF[n * PITCH + d];
        outb[n * DDIM + d] = f * c[r] + qacc[nt][r];
      }
    }
  }
}

extern "C" void kernel_launch(void* const* d_in, const int* in_sizes, int n_in,
                              void* d_out, int out_size, void* d_ws, size_t ws_size,
                              hipStream_t stream) {
  const float* F  = (const float*)d_in[0];
  const float* Kw = (const float*)d_in[1];
  const float* Qw = (const float*)d_in[2];
  float* out = (float*)d_out;
  const int B = in_sizes[0] / (NROWS * DDIM);  // 8192
  SAM3E_27934467293448_kernel<<<B, 128, 0, stream>>>(F, Kw, Qw, out);
}